// BatchedTGCN_6262062318297
// MI455X (gfx1250) — compile-verified
//
#include <hip/hip_runtime.h>
#include <hip/hip_bf16.h>

// ---------------- problem constants (from reference) ----------------
#define NN    10000       // nodes
#define NE    320000      // edges
#define FIN   2
#define HID   64
#define OUTD  16
#define BB    4
#define TT    12
#define NBR   (NN*BB)     // 40000 rows in flattened [N,B] layout
#define NBH   (NBR*HID)   // 2,560,000 floats per state tensor
#define NTILES (NBR/16)   // 2500 row-tiles of 16
#define NPACK (3*4*32*32) // packed gate-weight fragments (v2f each)
#define NPACKL (32*16)    // packed lin_w fragments (v2f each)

typedef __attribute__((ext_vector_type(2))) float v2f;
typedef __attribute__((ext_vector_type(8))) float v8f;

__device__ __forceinline__ v8f wmma4(v2f a, v2f b, v8f c) {
    // V_WMMA_F32_16X16X4_F32 : D = A(16x4) x B(4x16) + C(16x16), all f32
    return __builtin_amdgcn_wmma_f32_16x16x4_f32(
        false, a, false, b, (short)0, c, false, false);
}

__device__ __forceinline__ v2f ldg2(const float* p) {
    return *(const v2f*)p;   // 8B-aligned by construction
}

__device__ __forceinline__ float sigmoidf_(float x) {
    return 1.0f / (1.0f + expf(-x));
}

// One gate GEMM column-tile with pre-swizzled weights:
// D(16x16) = [C_in | H_in](16x128) @ W_col(128x16) + bias.
// bp points at this lane's 32 contiguous v2f B-fragments (compiler merges
// them into global_load_b128 clauses -> one wait per 32 WMMAs).
__device__ __forceinline__ v8f gate_tile(const v2f* aC, const v2f* aH,
                                         const v2f* __restrict__ bp, float bias) {
    v8f acc;
    #pragma unroll
    for (int j = 0; j < 8; ++j) acc[j] = bias;
    #pragma unroll
    for (int ks = 0; ks < 16; ++ks) acc = wmma4(aC[ks], bp[ks], acc);
    #pragma unroll
    for (int ks = 0; ks < 16; ++ks) acc = wmma4(aH[ks], bp[16 + ks], acc);
    return acc;
}

// ---------------- one-shot: pack weights into B-fragment order ----------------
// Bpack[((g*4+ct)*32 + lane)*32 + ks] = (W[krow][col], W[krow+1][col])
//   g: 0=z 1=r 2=h ; col = ct*16 + lane%16 ; kh = (lane/16)*2
//   krow = ks*4+kh (ks<16)  or  64+(ks-16)*4+kh (ks>=16)
__global__ void pack_w_kernel(const float* __restrict__ Lzw, const float* __restrict__ Lrw,
                              const float* __restrict__ Lhw, const float* __restrict__ linw,
                              v2f* __restrict__ Bpack, v2f* __restrict__ Lpack) {
    int idx = blockIdx.x * blockDim.x + threadIdx.x;
    if (idx < NPACK) {
        int ks   = idx & 31;
        int lane = (idx >> 5) & 31;
        int ct   = (idx >> 10) & 3;
        int g    = idx >> 12;
        int mlo  = lane & 15;
        int kh   = (lane >> 4) << 1;
        int krow = (ks < 16) ? (ks * 4 + kh) : (HID + (ks - 16) * 4 + kh);
        const float* Lw = (g == 0) ? Lzw : ((g == 1) ? Lrw : Lhw);
        int col = ct * 16 + mlo;
        v2f p;
        p.x = Lw[krow * HID + col];
        p.y = Lw[(krow + 1) * HID + col];
        Bpack[idx] = p;
    } else if (idx < NPACK + NPACKL) {
        int j    = idx - NPACK;
        int ks   = j & 15;
        int lane = j >> 4;
        int mlo  = lane & 15;
        int kh   = (lane >> 4) << 1;
        int krow = ks * 4 + kh;
        v2f p;
        p.x = linw[krow * OUTD + mlo];
        p.y = linw[(krow + 1) * OUTD + mlo];
        Lpack[j] = p;
    }
}

// ---------------- normalization precompute ----------------
__global__ void deg_init_kernel(float* __restrict__ deg) {
    int n = blockIdx.x * blockDim.x + threadIdx.x;
    if (n < NN) deg[n] = 1.0f;   // self-loop weight
}

__global__ void deg_acc_kernel(const int* __restrict__ dst,
                               const float* __restrict__ w,
                               float* __restrict__ deg) {
    int e = blockIdx.x * blockDim.x + threadIdx.x;
    if (e < NE) atomicAdd(&deg[dst[e]], w[e]);
}

__global__ void dinv_kernel(const float* __restrict__ deg,
                            float* __restrict__ dinv,
                            float* __restrict__ norm /* self-loop norm at [NE..NE+NN) */) {
    int n = blockIdx.x * blockDim.x + threadIdx.x;
    if (n >= NN) return;
    float d = deg[n];
    float dv = (d > 0.0f) ? rsqrtf(d) : 0.0f;
    dinv[n] = dv;
    norm[NE + n] = dv * dv;          // src==dst==n, w==1
}

__global__ void enorm_kernel(const int* __restrict__ src,
                             const int* __restrict__ dst,
                             const float* __restrict__ w,
                             const float* __restrict__ dinv,
                             float* __restrict__ norm) {
    int e = blockIdx.x * blockDim.x + threadIdx.x;
    if (e < NE) norm[e] = dinv[src[e]] * w[e] * dinv[dst[e]];
}

// ---------------- per-step: x @ W (K=2) for 3 gates + bias init of c ----------------
__global__ void xw_kernel(const float* __restrict__ x,
                          const float* __restrict__ Wz, const float* __restrict__ bz,
                          const float* __restrict__ Wr, const float* __restrict__ br,
                          const float* __restrict__ Wh, const float* __restrict__ bh,
                          float* __restrict__ xwz, float* __restrict__ xwr, float* __restrict__ xwh,
                          float* __restrict__ cz,  float* __restrict__ cr,  float* __restrict__ ch,
                          int t) {
    int idx = blockIdx.x * blockDim.x + threadIdx.x;   // over NBR*HID
    if (idx >= NBH) return;
    int h  = idx & (HID - 1);
    int nb = idx >> 6;
    int b  = nb & (BB - 1);
    int n  = nb >> 2;
    // x layout: [B, N, FIN, T]
    size_t base = (((size_t)b * NN + n) * FIN) * TT + t;
    float x0 = x[base];
    float x1 = x[base + TT];
    xwz[idx] = x0 * Wz[h] + x1 * Wz[HID + h];
    xwr[idx] = x0 * Wr[h] + x1 * Wr[HID + h];
    xwh[idx] = x0 * Wh[h] + x1 * Wh[HID + h];
    cz[idx] = bz[h];
    cr[idx] = br[h];
    ch[idx] = bh[h];
}

// ---------------- per-step: edge scatter for all 3 gates (L2-resident) ----------------
__global__ void scatter_kernel(const int* __restrict__ src, const int* __restrict__ dst,
                               const float* __restrict__ norm,
                               const float* __restrict__ xwz, const float* __restrict__ xwr,
                               const float* __restrict__ xwh,
                               float* __restrict__ cz, float* __restrict__ cr,
                               float* __restrict__ ch) {
    int idx = blockIdx.x * blockDim.x + threadIdx.x;   // over (NE+NN)*HID
    if (idx >= (NE + NN) * HID) return;
    int h = idx & (HID - 1);
    int e = idx >> 6;
    float nm = norm[e];
    int s, d;
    if (e < NE) { s = src[e]; d = dst[e]; }
    else        { s = d = e - NE; }
    #pragma unroll
    for (int b = 0; b < BB; ++b) {
        int si = (s * BB + b) * HID + h;
        int di = (d * BB + b) * HID + h;
        atomicAdd(&cz[di], xwz[si] * nm);
        atomicAdd(&cr[di], xwr[si] * nm);
        atomicAdd(&ch[di], xwh[si] * nm);
    }
}

// ---------------- per-step: GRU update + output projection (WMMA f32) ----------------
// One wave32 per 16-row tile. A-layout (16x4 f32): lanes 0-15 hold K={0,1},
// lanes 16-31 hold K={2,3}. D-layout (16x16 f32): VGPR j holds row M=j (lanes 0-15)
// and M=j+8 (lanes 16-31), N = lane%16.
// A-fragments live in registers; weights come from the pre-swizzled Bpack stream;
// H / H*R / relu(Hn) tiles round-trip via LDS.
__global__ __launch_bounds__(128)
void gru_step_kernel(const float* __restrict__ cz, const float* __restrict__ cr,
                     const float* __restrict__ ch,
                     const float* __restrict__ Hcur, float* __restrict__ Hnext,
                     const v2f* __restrict__ Bpack, const v2f* __restrict__ Lpack,
                     const float* __restrict__ Lzb, const float* __restrict__ Lrb,
                     const float* __restrict__ Lhb, const float* __restrict__ linb,
                     float* __restrict__ out, int t) {
    __shared__ float shH[4][16][HID];   // H tile (written A-layout, read D-layout)
    __shared__ float shHR[4][16][HID];  // H*R tile (written D-layout, read A-layout)
    __shared__ float shHN[4][16][HID];  // relu(Hn) tile

    const int wv   = threadIdx.x >> 5;
    const int lane = threadIdx.x & 31;
    const int tile = blockIdx.x * 4 + wv;
    if (tile >= NTILES) return;         // wave-uniform: EXEC stays all-ones

    const int r0   = tile * 16;
    const int mlo  = lane & 15;
    const int hi   = lane >> 4;         // 0 or 1
    const int kh   = hi << 1;           // K offset 0 or 2
    const int rowA = r0 + mlo;

    // ---- load H A-fragments once; stage full H tile into LDS --------------
    v2f aH[16];
    {
        const float* hA = Hcur + rowA * HID + kh;
        #pragma unroll
        for (int ks = 0; ks < 16; ++ks) aH[ks] = ldg2(hA + ks * 4);
        #pragma unroll
        for (int ks = 0; ks < 16; ++ks)
            *(v2f*)&shH[wv][mlo][ks * 4 + kh] = aH[ks];
    }

    // ---- R gate: sigmoid([cr,H] @ Lrw + Lrb)  (g=1 in Bpack) --------------
    v2f aC[16];
    {
        const float* cA = cr + rowA * HID + kh;
        #pragma unroll
        for (int ks = 0; ks < 16; ++ks) aC[ks] = ldg2(cA + ks * 4);
    }
    v8f rreg[4];
    #pragma unroll
    for (int ct = 0; ct < 4; ++ct) {
        const v2f* bp = Bpack + ((size_t)(1 * 4 + ct) * 32 + lane) * 32;
        v8f acc = gate_tile(aC, aH, bp, Lrb[ct * 16 + mlo]);
        #pragma unroll
        for (int j = 0; j < 8; ++j) acc[j] = sigmoidf_(acc[j]);
        rreg[ct] = acc;
    }

    // ---- H*R tile (D-layout reads of H from LDS) --------------------------
    #pragma unroll
    for (int ct = 0; ct < 4; ++ct) {
        int col = ct * 16 + mlo;
        #pragma unroll
        for (int j = 0; j < 8; ++j) {
            int mj = j + hi * 8;
            shHR[wv][mj][col] = shH[wv][mj][col] * rreg[ct][j];
        }
    }

    // ---- Z gate: sigmoid([cz,H] @ Lzw + Lzb)  (g=0 in Bpack) --------------
    {
        const float* cA = cz + rowA * HID + kh;
        #pragma unroll
        for (int ks = 0; ks < 16; ++ks) aC[ks] = ldg2(cA + ks * 4);
    }
    v8f zreg[4];
    #pragma unroll
    for (int ct = 0; ct < 4; ++ct) {
        const v2f* bp = Bpack + ((size_t)(0 * 4 + ct) * 32 + lane) * 32;
        v8f acc = gate_tile(aC, aH, bp, Lzb[ct * 16 + mlo]);
        #pragma unroll
        for (int j = 0; j < 8; ++j) acc[j] = sigmoidf_(acc[j]);
        zreg[ct] = acc;
    }

    // ---- H~ = tanh([ch, H*R] @ Lhw + Lhb); GRU combine  (g=2) -------------
    v2f aHR[16];
    {
        const float* cA = ch + rowA * HID + kh;
        #pragma unroll
        for (int ks = 0; ks < 16; ++ks) aC[ks] = ldg2(cA + ks * 4);
        #pragma unroll
        for (int ks = 0; ks < 16; ++ks)
            aHR[ks] = *(const v2f*)&shHR[wv][mlo][ks * 4 + kh];
    }
    #pragma unroll
    for (int ct = 0; ct < 4; ++ct) {
        const v2f* bp = Bpack + ((size_t)(2 * 4 + ct) * 32 + lane) * 32;
        v8f acc = gate_tile(aC, aHR, bp, Lhb[ct * 16 + mlo]);
        int col = ct * 16 + mlo;
        #pragma unroll
        for (int j = 0; j < 8; ++j) {
            float htl = tanhf(acc[j]);
            float z   = zreg[ct][j];
            int   mj  = j + hi * 8;
            float hv  = shH[wv][mj][col];
            float hn  = z * hv + (1.0f - z) * htl;
            Hnext[(r0 + mj) * HID + col] = hn;
            shHN[wv][mj][col] = fmaxf(hn, 0.0f);
        }
    }

    // ---- output projection: relu(Hn) @ lin_w + lin_b (64 -> 16) -----------
    {
        v2f aO[16];
        #pragma unroll
        for (int ks = 0; ks < 16; ++ks)
            aO[ks] = *(const v2f*)&shHN[wv][mlo][ks * 4 + kh];
        float bias = linb[mlo];
        v8f acc;
        #pragma unroll
        for (int j = 0; j < 8; ++j) acc[j] = bias;
        const v2f* bl = Lpack + lane * 16;
        #pragma unroll
        for (int ks = 0; ks < 16; ++ks) acc = wmma4(aO[ks], bl[ks], acc);
        #pragma unroll
        for (int j = 0; j < 8; ++j) {
            int mj  = j + hi * 8;
            int row = r0 + mj;
            int n   = row >> 2;                    // row = n*B + b
            int b   = row & 3;
            out[(((size_t)b * TT + t) * NN + n) * OUTD + mlo] = acc[j];
        }
    }
}

// ---------------- host launch ----------------
extern "C" void kernel_launch(void* const* d_in, const int* in_sizes, int n_in,
                              void* d_out, int out_size, void* d_ws, size_t ws_size,
                              hipStream_t stream) {
    const float* x    = (const float*)d_in[0];
    const int*   eidx = (const int*)d_in[1];
    const float* ew   = (const float*)d_in[2];
    const float* Wz   = (const float*)d_in[3];  const float* bz  = (const float*)d_in[4];
    const float* Wr   = (const float*)d_in[5];  const float* br  = (const float*)d_in[6];
    const float* Wh   = (const float*)d_in[7];  const float* bh  = (const float*)d_in[8];
    const float* Lzw  = (const float*)d_in[9];  const float* Lzb = (const float*)d_in[10];
    const float* Lrw  = (const float*)d_in[11]; const float* Lrb = (const float*)d_in[12];
    const float* Lhw  = (const float*)d_in[13]; const float* Lhb = (const float*)d_in[14];
    const float* linw = (const float*)d_in[15]; const float* linb = (const float*)d_in[16];
    float* out = (float*)d_out;

    // workspace carve-up (~83.4 MB total)
    float* ws   = (float*)d_ws;
    float* deg  = ws;                       // NN
    float* dinv = deg  + NN;                // NN
    float* norm = dinv + NN;                // NE + NN   (350000 floats total: 16B aligned)
    float* xwz  = norm + (NE + NN);         // NBH each
    float* xwr  = xwz + NBH;
    float* xwh  = xwr + NBH;
    float* czb  = xwh + NBH;
    float* crb  = czb + NBH;
    float* chb  = crb + NBH;
    float* Ha   = chb + NBH;
    float* Hb   = Ha  + NBH;
    v2f*   Bpack = (v2f*)(Hb + NBH);        // NPACK v2f (16B-aligned offset)
    v2f*   Lpack = Bpack + NPACK;           // NPACKL v2f

    const int* src = eidx;
    const int* dst = eidx + NE;

    // one-shot precompute: normalization + weight swizzle
    deg_init_kernel<<<(NN + 255) / 256, 256, 0, stream>>>(deg);
    deg_acc_kernel<<<(NE + 255) / 256, 256, 0, stream>>>(dst, ew, deg);
    dinv_kernel<<<(NN + 255) / 256, 256, 0, stream>>>(deg, dinv, norm);
    enorm_kernel<<<(NE + 255) / 256, 256, 0, stream>>>(src, dst, ew, dinv, norm);
    pack_w_kernel<<<(NPACK + NPACKL + 255) / 256, 256, 0, stream>>>(
        Lzw, Lrw, Lhw, linw, Bpack, Lpack);

    // H0 = 0
    hipMemsetAsync(Ha, 0, (size_t)NBH * sizeof(float), stream);

    float* Hc = Ha;
    float* Hn = Hb;
    const int scatter_threads = (NE + NN) * HID;
    for (int t = 0; t < TT; ++t) {
        xw_kernel<<<(NBH + 255) / 256, 256, 0, stream>>>(
            x, Wz, bz, Wr, br, Wh, bh, xwz, xwr, xwh, czb, crb, chb, t);
        scatter_kernel<<<(scatter_threads + 255) / 256, 256, 0, stream>>>(
            src, dst, norm, xwz, xwr, xwh, czb, crb, chb);
        gru_step_kernel<<<(NTILES + 3) / 4, 128, 0, stream>>>(
            czb, crb, chb, Hc, Hn, Bpack, Lpack, Lzb, Lrb, Lhb, linb, out, t);
        float* tmp = Hc; Hc = Hn; Hn = tmp;
    }
}